// Memory_30537217475134
// MI455X (gfx1250) — compile-verified
//
#include <hip/hip_runtime.h>
#include <math.h>

// NTM memory addressing step: B=8192, N=128, M=64, fp32.
// Memory-bound (~517MB total HBM traffic -> ~22us floor at 23.3TB/s).
// One 256-thread block per batch; 128x64 tile staged once in LDS.
// Matrix ops (dot = mem@k, read = w@mem) use V_WMMA_F32_16X16X4_F32 (full fp32).

#define EPSV 1e-8f
#define NB 8192
#define NN 128
#define MM 64
#define MSTR 68  // padded LDS row stride (floats): stride-4 banks -> conflict-free column access

typedef __attribute__((ext_vector_type(2))) float v2f;
typedef __attribute__((ext_vector_type(4))) float v4f;
typedef __attribute__((ext_vector_type(8))) float v8f;

__device__ __forceinline__ float blk_reduce_sum(float* red, int tid, float v) {
    red[tid] = v;
    __syncthreads();
#pragma unroll
    for (int s = 128; s > 0; s >>= 1) {
        if (tid < s) red[tid] += red[tid + s];
        __syncthreads();
    }
    float r = red[0];
    __syncthreads();
    return r;
}

__device__ __forceinline__ float blk_reduce_max(float* red, int tid, float v) {
    red[tid] = v;
    __syncthreads();
#pragma unroll
    for (int s = 128; s > 0; s >>= 1) {
        if (tid < s) red[tid] = fmaxf(red[tid], red[tid + s]);
        __syncthreads();
    }
    float r = red[0];
    __syncthreads();
    return r;
}

__global__ __launch_bounds__(256) void ntm_step_kernel(
    const float* __restrict__ memory, const float* __restrict__ prev_w,
    const float* __restrict__ k, const float* __restrict__ beta,
    const float* __restrict__ g, const float* __restrict__ s,
    const float* __restrict__ y, const float* __restrict__ e,
    const float* __restrict__ a,
    float* __restrict__ out_read, float* __restrict__ out_mem,
    float* __restrict__ out_w)
{
    __shared__ float lds_mem[NN * MSTR];   // 34816 B
    __shared__ float lds_k[MM];
    __shared__ float lds_e[MM];
    __shared__ float lds_a[MM];
    __shared__ float lds_dot[NN];
    __shared__ float lds_mnorm[NN];
    __shared__ float lds_wg[NN];
    __shared__ float lds_w[NN];
    __shared__ float lds_part[256];
    __shared__ float lds_red[256];
    __shared__ float lds_pread[2][MM];

    const int tid  = threadIdx.x;
    const int lane = tid & 31;
    const int wv   = tid >> 5;
    const int b    = blockIdx.x;

    const float* memb = memory + (size_t)b * (NN * MM);

    // ---- phase 1: stream memory tile to LDS (coalesced, non-temporal) + k/e/a ----
    {
        const v4f* gm = (const v4f*)memb;
#pragma unroll
        for (int i = 0; i < 8; ++i) {
            int f   = i * 256 + tid;      // 0..2047 float4s
            int row = f >> 4;             // 16 float4 per 64-float row
            int c4  = f & 15;
            v4f v = __builtin_nontemporal_load(&gm[f]);
            *(v4f*)&lds_mem[row * MSTR + c4 * 4] = v;  // row*68 %4==0 -> 16B aligned
        }
        if (tid < MM) {
            lds_k[tid] = k[(size_t)b * MM + tid];
            lds_e[tid] = e[(size_t)b * MM + tid];
            lds_a[tid] = a[(size_t)b * MM + tid];
        }
    }
    __syncthreads();

    const float beta_b = beta[b];
    const float g_b    = g[b];
    const float y_b    = y[b];
    const float s0 = s[0], s1 = s[1], s2 = s[2];

    // ---- phase 2: row sum-of-squares (2 threads/row) + ||k|| ----
    {
        int row = tid >> 1, h = tid & 1;
        const float* r = &lds_mem[row * MSTR + h * 32];
        float ss = 0.f;
#pragma unroll
        for (int c = 0; c < 32; ++c) { float v = r[c]; ss += v * v; }
        lds_part[tid] = ss;
    }
    float kq = (tid < MM) ? lds_k[tid] * lds_k[tid] : 0.f;
    float ksum = blk_reduce_sum(lds_red, tid, kq);       // barriers make lds_part visible
    const float knorm = fmaxf(sqrtf(ksum), EPSV);
    if (tid < NN)
        lds_mnorm[tid] = fmaxf(sqrtf(lds_part[2 * tid] + lds_part[2 * tid + 1]), EPSV);

    // ---- phase 3: dot[n] = sum_m mem[n,m]*k[m] via f32 WMMA ----
    // A = 16-row tile of memory (16x4 per step), B = k chunk replicated over 16 cols.
    // K -> (vgpr, lanehalf) mapping: K = 2*lanehalf + vgpr (per documented A layout,
    // mirrored for B). Full EXEC here (no divergence).
    {
        v8f acc = {0.f, 0.f, 0.f, 0.f, 0.f, 0.f, 0.f, 0.f};
        const int row   = (wv << 4) + (lane & 15);
        const int khalf = (lane >> 4) << 1;  // 0 or 2
        const float* mrow = &lds_mem[row * MSTR];
#pragma unroll
        for (int k0 = 0; k0 < MM; k0 += 4) {
            v2f av, bv;
            av.x = mrow[k0 + khalf];
            av.y = mrow[k0 + khalf + 1];
            bv.x = lds_k[k0 + khalf];
            bv.y = lds_k[k0 + khalf + 1];
            acc = __builtin_amdgcn_wmma_f32_16x16x4_f32(false, av, false, bv,
                                                        (short)0, acc, false, false);
        }
        if ((lane & 15) == 0) {  // D[i][j] is j-independent; lanes 0/16 hold rows n0+0..7 / +8..15
            int base = (wv << 4) + ((lane >> 4) << 3);
#pragma unroll
            for (int v = 0; v < 8; ++v) lds_dot[base + v] = acc[v];
        }
    }
    __syncthreads();

    // ---- phase 4: softmax(beta*cos) -> interpolate -> shift -> sharpen ----
    float logit = -INFINITY;
    if (tid < NN) {
        float cosv = lds_dot[tid] / (knorm * lds_mnorm[tid]);
        logit = beta_b * cosv;
    }
    float mx = blk_reduce_max(lds_red, tid, logit);
    float ex = (tid < NN) ? expf(logit - mx) : 0.f;
    float se = blk_reduce_sum(lds_red, tid, ex);
    if (tid < NN) {
        float wc  = ex / se;
        float wgv = g_b * wc + (1.f - g_b) * prev_w[(size_t)b * NN + tid];
        lds_wg[tid] = wgv;
    }
    __syncthreads();
    float wp = 0.f;
    if (tid < NN) {
        float wt = s0 * lds_wg[(tid + NN - 1) & (NN - 1)] + s1 * lds_wg[tid] +
                   s2 * lds_wg[(tid + 1) & (NN - 1)];
        wp = powf(wt, y_b);
    }
    float sw = blk_reduce_sum(lds_red, tid, wp);
    if (tid < NN) {
        float wf = wp / sw;
        lds_w[tid] = wf;
        out_w[(size_t)b * NN + tid] = wf;
    }
    __syncthreads();

    // ---- phase 5: read[m] = sum_n w[n]*mem[n,m] via f32 WMMA ----
    // 8 waves: 4 column tiles x 2 K-halves (16 WMMAs each), partials summed in LDS.
    {
        v8f acc = {0.f, 0.f, 0.f, 0.f, 0.f, 0.f, 0.f, 0.f};
        const int j0    = (wv & 3) << 4;
        const int kb    = (wv >> 2) << 6;   // 0 or 64
        const int khalf = (lane >> 4) << 1; // 0 or 2
        const int j     = lane & 15;
#pragma unroll
        for (int k0 = 0; k0 < 64; k0 += 4) {
            int kk = kb + k0 + khalf;
            v2f av, bv;
            av.x = lds_w[kk];                       // A rows replicated with w
            av.y = lds_w[kk + 1];
            bv.x = lds_mem[kk * MSTR + j0 + j];     // B = memory chunk (K x 16 cols)
            bv.y = lds_mem[(kk + 1) * MSTR + j0 + j];
            acc = __builtin_amdgcn_wmma_f32_16x16x4_f32(false, av, false, bv,
                                                        (short)0, acc, false, false);
        }
        if (lane < 16) lds_pread[wv >> 2][j0 + j] = acc[0];  // D row 0 (rows identical)
    }
    __syncthreads();
    if (tid < MM)
        out_read[(size_t)b * MM + tid] = lds_pread[0][tid] + lds_pread[1][tid];

    // ---- phase 6: mem_new = mem*(1 - w⊗e) + w⊗a, streamed out non-temporal ----
    {
        v4f* om = (v4f*)(out_mem + (size_t)b * (NN * MM));
#pragma unroll
        for (int i = 0; i < 8; ++i) {
            int f   = i * 256 + tid;
            int row = f >> 4;
            int c4  = f & 15;
            float wr = lds_w[row];
            v4f m = *(const v4f*)&lds_mem[row * MSTR + c4 * 4];
            int c = c4 * 4;
            v4f o;
            o.x = m.x * (1.f - wr * lds_e[c + 0]) + wr * lds_a[c + 0];
            o.y = m.y * (1.f - wr * lds_e[c + 1]) + wr * lds_a[c + 1];
            o.z = m.z * (1.f - wr * lds_e[c + 2]) + wr * lds_a[c + 2];
            o.w = m.w * (1.f - wr * lds_e[c + 3]) + wr * lds_a[c + 3];
            __builtin_nontemporal_store(o, &om[f]);
        }
    }
}

extern "C" void kernel_launch(void* const* d_in, const int* in_sizes, int n_in,
                              void* d_out, int out_size, void* d_ws, size_t ws_size,
                              hipStream_t stream) {
    (void)in_sizes; (void)n_in; (void)d_ws; (void)ws_size; (void)out_size;
    const float* memory = (const float*)d_in[0];
    const float* prev_w = (const float*)d_in[1];
    const float* k      = (const float*)d_in[2];
    const float* beta   = (const float*)d_in[3];
    const float* g      = (const float*)d_in[4];
    const float* s      = (const float*)d_in[5];
    const float* y      = (const float*)d_in[6];
    const float* e      = (const float*)d_in[7];
    const float* a      = (const float*)d_in[8];

    float* out_read = (float*)d_out;                         // (B, M)
    float* out_mem  = out_read + (size_t)NB * MM;            // (B, N, M)
    float* out_w    = out_mem + (size_t)NB * NN * MM;        // (B, N)

    ntm_step_kernel<<<dim3(NB), dim3(256), 0, stream>>>(
        memory, prev_w, k, beta, g, s, y, e, a, out_read, out_mem, out_w);
}